// onering_conv_layer_batch_17557826306183
// MI455X (gfx1250) — compile-verified
//
#include <hip/hip_runtime.h>
#include <hip/hip_bf16.h>

typedef __attribute__((ext_vector_type(2))) float v2f;
typedef __attribute__((ext_vector_type(8))) float v8f;

#define KNB 7
#define CCH 64
#define OUTCH 64
#define WSTRIDE 450   // 448 + 2 pad: keeps ds_load_b64 8B-aligned, kills bank conflicts

// ---------------------------------------------------------------------------
// Pass 1: transpose x [B, C, N] -> x_t [B, N, C] so gathers become contiguous
// 256-byte row reads.  Classic 32x32 LDS tile transpose, coalesced both ways.
// ---------------------------------------------------------------------------
__global__ void xt_transpose_kernel(const float* __restrict__ x,
                                    float* __restrict__ xt, int N) {
    __shared__ float tile[32][33];
    const int b  = blockIdx.z;
    const int n0 = blockIdx.x * 32;
    const int c0 = blockIdx.y * 32;
    const int tx = threadIdx.x;        // 0..31
    const int ty = threadIdx.y;        // 0..7
    const float* xb  = x  + (size_t)b * CCH * N;
    float*       xtb = xt + (size_t)b * N * CCH;
#pragma unroll
    for (int i = ty; i < 32; i += 8) {
        const int c = c0 + i, n = n0 + tx;
        tile[i][tx] = (n < N) ? xb[(size_t)c * N + n] : 0.0f;
    }
    __syncthreads();
#pragma unroll
    for (int i = ty; i < 32; i += 8) {
        const int n = n0 + i, c = c0 + tx;
        if (n < N) xtb[(size_t)n * CCH + c] = tile[tx][i];
    }
}

// ---------------------------------------------------------------------------
// Pass 2: fused gather + GEMM via V_WMMA_F32_16X16X4_F32.
// One wave = 16 rows (n) x 64 outputs (4 accumulator tiles).
// 8 waves / block share the LDS-resident 112KB weight matrix.
// XT=true  : x_t [B,N,C] layout -> A fragment is one global_load_b64/lane.
// XT=false : original [B,C,N] layout fallback (ws too small), strided loads.
// ---------------------------------------------------------------------------
template <bool XT>
__global__ void gather_wmma_kernel(const float* __restrict__ xsrc,
                                   const long long* __restrict__ nidx,
                                   const float* __restrict__ W,
                                   const float* __restrict__ bias,
                                   float* __restrict__ out, int N) {
    __shared__ float lds_w[OUTCH * WSTRIDE];   // 115,200 B of the 320KB WGP LDS

    const int tid = threadIdx.x;               // 0..255 (8 wave32s)

    // --- cooperative W load: 64 x 448 floats, float4 global, v2f LDS stores
    {
        const float4* Wv = (const float4*)W;
#pragma unroll 4
        for (int i = tid; i < (OUTCH * KNB * CCH) / 4; i += 256) {
            float4 v = Wv[i];
            const int flat = i * 4;
            const int o = flat / (KNB * CCH);
            const int kc = flat - o * (KNB * CCH);
            float* dst = &lds_w[o * WSTRIDE + kc];  // 8B aligned (even dword idx)
            v2f lo; lo.x = v.x; lo.y = v.y;
            v2f hi; hi.x = v.z; hi.y = v.w;
            *(v2f*)(dst)     = lo;
            *(v2f*)(dst + 2) = hi;
        }
    }
    __syncthreads();

    const int lane = tid & 31;
    const int wave = tid >> 5;                 // 0..7
    const int b    = blockIdx.y;
    const int n0   = blockIdx.x * 128 + wave * 16;
    if (n0 >= N) return;                       // idle tail waves (after barrier)

    const int  r    = lane & 15;               // tile row / output column slot
    const bool hiH  = (lane >= 16);
    const int  koff = hiH ? 2 : 0;             // ISA A/B frag: hi half holds K+2,K+3

    // per-lane neighbor indices for this lane's row (clamped for partial tiles)
    int nr = n0 + r; if (nr >= N) nr = N - 1;
    int j[KNB];
#pragma unroll
    for (int k = 0; k < KNB; ++k) j[k] = (int)nidx[(size_t)nr * KNB + k];

    const float* xb = xsrc + (size_t)b * N * CCH;   // same flat size both layouts

    v8f acc0 = {}, acc1 = {}, acc2 = {}, acc3 = {};

#pragma unroll
    for (int k = 0; k < KNB; ++k) {
        const int jj = j[k];
#pragma unroll
        for (int c4 = 0; c4 < CCH / 4; ++c4) {
            const int c0    = c4 * 4 + koff;        // this lane's K-pair start
            const int kappa = k * CCH + c0;         // flattened K index into W

            // A fragment: x_t[b][jj][c0..c0+1]  (or strided from [C,N] layout)
            v2f a;
            if (XT) {
                a = *(const v2f*)(xb + (size_t)jj * CCH + c0);
            } else {
                a.x = xb[(size_t)c0 * N + jj];
                a.y = xb[(size_t)(c0 + 1) * N + jj];
            }

            // B fragments: W[o][kappa..kappa+1] straight out of LDS rows
            v2f b0 = *(const v2f*)(&lds_w[( 0 + r) * WSTRIDE + kappa]);
            v2f b1 = *(const v2f*)(&lds_w[(16 + r) * WSTRIDE + kappa]);
            v2f b2 = *(const v2f*)(&lds_w[(32 + r) * WSTRIDE + kappa]);
            v2f b3 = *(const v2f*)(&lds_w[(48 + r) * WSTRIDE + kappa]);

            acc0 = __builtin_amdgcn_wmma_f32_16x16x4_f32(false, a, false, b0,
                                                         (short)0, acc0, false, false);
            acc1 = __builtin_amdgcn_wmma_f32_16x16x4_f32(false, a, false, b1,
                                                         (short)0, acc1, false, false);
            acc2 = __builtin_amdgcn_wmma_f32_16x16x4_f32(false, a, false, b2,
                                                         (short)0, acc2, false, false);
            acc3 = __builtin_amdgcn_wmma_f32_16x16x4_f32(false, a, false, b3,
                                                         (short)0, acc3, false, false);
        }
    }

    // --- epilogue: D[i] holds row (i + hi*8), col (lane&15); add bias, store
    const float bs0 = bias[ 0 + r];
    const float bs1 = bias[16 + r];
    const float bs2 = bias[32 + r];
    const float bs3 = bias[48 + r];
    float* ob = out + (size_t)b * OUTCH * N;
#pragma unroll
    for (int i = 0; i < 8; ++i) {
        const int m = i + (hiH ? 8 : 0);
        const int n = n0 + m;
        if (n < N) {
            ob[(size_t)( 0 + r) * N + n] = acc0[i] + bs0;
            ob[(size_t)(16 + r) * N + n] = acc1[i] + bs1;
            ob[(size_t)(32 + r) * N + n] = acc2[i] + bs2;
            ob[(size_t)(48 + r) * N + n] = acc3[i] + bs3;
        }
    }
}

extern "C" void kernel_launch(void* const* d_in, const int* in_sizes, int n_in,
                              void* d_out, int out_size, void* d_ws, size_t ws_size,
                              hipStream_t stream) {
    const float*     x    = (const float*)d_in[0];
    const long long* nidx = (const long long*)d_in[1];   // jax int64
    const float*     W    = (const float*)d_in[2];
    const float*     bias = (const float*)d_in[3];
    float*           out  = (float*)d_out;

    const int B = 8, N = 40962;
    const size_t xt_bytes = (size_t)B * N * CCH * sizeof(float);

    dim3 g((N + 127) / 128, B);
    if (ws_size >= xt_bytes) {
        float* xt = (float*)d_ws;
        dim3 tb(32, 8);
        dim3 tg((N + 31) / 32, CCH / 32, B);
        xt_transpose_kernel<<<tg, tb, 0, stream>>>(x, xt, N);
        gather_wmma_kernel<true><<<g, 256, 0, stream>>>(xt, nidx, W, bias, out, N);
    } else {
        gather_wmma_kernel<false><<<g, 256, 0, stream>>>(x, nidx, W, bias, out, N);
    }
}